// CoCorrRec_23965917511737
// MI455X (gfx1250) — compile-verified
//
#include <hip/hip_runtime.h>
#include <stdint.h>

#ifndef __has_builtin
#define __has_builtin(x) 0
#endif
#if __has_builtin(__builtin_amdgcn_global_load_async_to_lds_b128)
#define HAS_ASYNC 1
#else
#define HAS_ASYNC 0
#endif

typedef __attribute__((ext_vector_type(2))) float v2f;
typedef __attribute__((ext_vector_type(8))) float v8f;
typedef int v4i_t __attribute__((vector_size(16)));   // matches builtin's int4 pointee

#define LDP 68   // padded row stride (dwords) for 16x64 tiles (272B = 17*16B)
#define LDW 68   // padded row stride for the 64x64 W1 state
#define LDE 20   // padded row stride for 16x16 eta tiles (80B = 5*16B)
#define LDM 17   // padded row stride for the 16x16 M tile (not async-loaded)

struct __align__(16) Smem {
  float W1[64 * LDW];       // persistent fp32 state
  float XQ[2][16 * LDP];    // double-buffered input tiles
  float XK[2][16 * LDP];
  float XV[2][16 * LDP];
  float Eta[2][16 * LDE];
  float G [16 * LDP];       // LN-bwd gradient g
  float Z [16 * LDP];       // Z1 staging
  float M [16 * LDM];       // -(eta * tril * (S+1))
  float b1[64];
  float gam[64];
  float bet[64];
  float mu[16];
  float rstd[16];
  float negEta[16];         // -eta[15][k]
  float P1[128];
  float P2[128];
  float RS[64];
  float RQ[64];
};

__device__ __forceinline__ v2f ldsA(const float* m, int ld, int k0, int n16, int kh) {
  // A 16x4 frag: lane row = n16, K = k0+2*kh (+1)
  const float* p = m + n16 * ld + k0 + 2 * kh;
  v2f a; a.x = p[0]; a.y = p[1]; return a;
}
__device__ __forceinline__ v2f ldsB(const float* m, int ld, int k0, int cb, int n16, int kh) {
  // B 4x16 frag: lane col = cb+n16, K = k0+2*kh (+1)
  const float* p = m + (k0 + 2 * kh) * ld + cb + n16;
  v2f b; b.x = p[0]; b.y = p[ld]; return b;
}
__device__ __forceinline__ v8f wmma4(v2f a, v2f b, v8f c) {
  return __builtin_amdgcn_wmma_f32_16x16x4_f32(false, a, false, b, (short)0, c, false, false);
}

// Sum across each 16-lane half of the wave (rows of the C fragment).
// v_permlane16_b32 keeps this entirely on the VALU (no ds_bpermute traffic).
__device__ __forceinline__ float redrow16(float v) {
#if __has_builtin(__builtin_amdgcn_permlane16)
  int i;
  i = __float_as_int(v);
  v += __int_as_float(__builtin_amdgcn_permlane16(i, i, 0x67452301u, 0xEFCDAB89u, false, false)); // xor 1
  i = __float_as_int(v);
  v += __int_as_float(__builtin_amdgcn_permlane16(i, i, 0x54761032u, 0xDCFE98BAu, false, false)); // xor 2
  i = __float_as_int(v);
  v += __int_as_float(__builtin_amdgcn_permlane16(i, i, 0x32107654u, 0xBA98FEDCu, false, false)); // xor 4
  i = __float_as_int(v);
  v += __int_as_float(__builtin_amdgcn_permlane16(i, i, 0xFEDCBA98u, 0x76543210u, false, false)); // xor 8
#else
  #pragma unroll
  for (int d = 1; d < 16; d <<= 1) v += __shfl_xor(v, d, 32);
#endif
  return v;
}

__device__ __forceinline__ void wait_async0() {
#if HAS_ASYNC
#if __has_builtin(__builtin_amdgcn_s_wait_asynccnt)
  __builtin_amdgcn_s_wait_asynccnt(0);
#else
  asm volatile("s_wait_asynccnt 0x0" ::: "memory");
#endif
#endif
}

#if HAS_ASYNC
#define GPTR(p) ((__attribute__((address_space(1))) v4i_t*)(p))
#define LPTR(p) ((__attribute__((address_space(3))) v4i_t*)(p))
// 16x64 fp32 tile = 256 x 16B chunks; 128 threads issue 2 each
__device__ __forceinline__ void issue_tile(const float* g, float* l, int tid) {
  int c = tid, r = c >> 4, q = (c & 15) << 2;
  __builtin_amdgcn_global_load_async_to_lds_b128(GPTR(g + r * 64 + q),
                                                 LPTR(l + r * LDP + q), 0, 0);
  c = tid + 128; r = c >> 4; q = (c & 15) << 2;
  __builtin_amdgcn_global_load_async_to_lds_b128(GPTR(g + r * 64 + q),
                                                 LPTR(l + r * LDP + q), 0, 0);
}
// 16x16 fp32 tile = 64 x 16B chunks; threads 0..63 issue 1 each
__device__ __forceinline__ void issue_eta(const float* g, float* l, int tid) {
  if (tid < 64) {
    int r = tid >> 2, q = (tid & 3) << 2;
    __builtin_amdgcn_global_load_async_to_lds_b128(GPTR(g + r * 16 + q),
                                                   LPTR(l + r * LDE + q), 0, 0);
  }
}
#endif

__global__ __launch_bounds__(128) void ttt_scan_kernel(
    const float* __restrict__ XQ, const float* __restrict__ XK,
    const float* __restrict__ XV, const float* __restrict__ eta,
    const float* __restrict__ W1g, const float* __restrict__ b1g,
    const float* __restrict__ lnw, const float* __restrict__ lnb,
    float* __restrict__ out)
{
  __shared__ Smem s;
  const int tid  = threadIdx.x;
  const int lane = tid & 31;
  const int wv   = tid >> 5;      // wave 0..3 -> owns cols [16*wv, 16*wv+16)
  const int n16  = lane & 15;
  const int kh   = lane >> 4;     // half-wave selector
  const int cb   = wv * 16;
  const int bh   = blockIdx.x;    // b*8 + h
  const int h    = bh & 7;
  const int b    = bh >> 3;

  const size_t chainX = (size_t)bh * 256 * 1024;  // floats
  const size_t chainE = (size_t)bh * 256 * 256;
  const float* gq0 = XQ  + chainX;
  const float* gk0 = XK  + chainX;
  const float* gv0 = XV  + chainX;
  const float* ge0 = eta + chainE;
  float* outB = out + (size_t)b * 4096 * 512 + (size_t)h * 64;

#if HAS_ASYNC
  // kick off step-0 tiles; DMA overlaps the state init below
  issue_tile(gq0, s.XQ[0], tid);
  issue_tile(gk0, s.XK[0], tid);
  issue_tile(gv0, s.XV[0], tid);
  issue_eta (ge0, s.Eta[0], tid);
#endif

  // ---- one-time state init ----
  for (int i = tid; i < 4096; i += 128)
    s.W1[(i >> 6) * LDW + (i & 63)] = W1g[h * 4096 + i];
  if (tid < 64) {
    s.b1[tid]  = b1g[h * 64 + tid];
    s.gam[tid] = lnw[h * 64 + tid];
    s.bet[tid] = lnb[h * 64 + tid];
  }

  for (int t = 0; t < 256; ++t) {
    const int cur = t & 1;
    float* xq = s.XQ[cur];
    float* xk = s.XK[cur];
    float* xv = s.XV[cur];
    float* et = s.Eta[cur];

#if HAS_ASYNC
    wait_async0();          // my async writes to LDS are done
    __syncthreads();        // B1: everyone's writes are done; prior step fully retired
    if (t + 1 < 256) {      // overlap next step's DMA with this step's compute
      issue_tile(gq0 + (size_t)(t + 1) * 1024, s.XQ[cur ^ 1], tid);
      issue_tile(gk0 + (size_t)(t + 1) * 1024, s.XK[cur ^ 1], tid);
      issue_tile(gv0 + (size_t)(t + 1) * 1024, s.XV[cur ^ 1], tid);
      issue_eta (ge0 + (size_t)(t + 1) * 256,  s.Eta[cur ^ 1], tid);
    }
#else
    {
      const float* gq = gq0 + (size_t)t * 1024;
      const float* gk = gk0 + (size_t)t * 1024;
      const float* gv = gv0 + (size_t)t * 1024;
      const float* ge = ge0 + (size_t)t * 256;
      for (int i = tid; i < 1024; i += 128) {
        int r = i >> 6, cc = i & 63;
        xq[r * LDP + cc] = gq[i];
        xk[r * LDP + cc] = gk[i];
        xv[r * LDP + cc] = gv[i];
      }
      for (int i = tid; i < 256; i += 128)
        et[(i >> 4) * LDE + (i & 15)] = ge[i];
      if (tid == 0 && t + 1 < 256) {
        __builtin_prefetch(gq + 1024, 0, 1);
        __builtin_prefetch(gk + 1024, 0, 1);
        __builtin_prefetch(gv + 1024, 0, 1);
        __builtin_prefetch(ge + 256, 0, 1);
      }
      __syncthreads();      // B1
    }
#endif

    // ---- Z1 = xk @ W1 + b1 (this wave's 16 cols) ----
    v8f c;
    {
      float bb = s.b1[cb + n16];
      #pragma unroll
      for (int i = 0; i < 8; ++i) c[i] = bb;
      #pragma unroll
      for (int kb = 0; kb < 16; ++kb)
        c = wmma4(ldsA(xk, LDP, kb * 4, n16, kh),
                  ldsB(s.W1, LDW, kb * 4, cb, n16, kh), c);
    }
    // stash Z1 + per-row partial sum/sumsq via VALU lane-permutes
    #pragma unroll
    for (int i = 0; i < 8; ++i) {
      int m = i + 8 * kh;
      s.Z[m * LDP + cb + n16] = c[i];
      float sv = redrow16(c[i]);
      float qv = redrow16(c[i] * c[i]);
      if (n16 == 0) { s.RS[wv * 16 + m] = sv; s.RQ[wv * 16 + m] = qv; }
    }
    // wave0 also computes S = xq @ xk^T and M = -(eta ∘ tril ∘ (S+1)) before B2
    if (wv == 0) {
      v8f sf;
      #pragma unroll
      for (int i = 0; i < 8; ++i) sf[i] = 0.f;
      #pragma unroll
      for (int kb = 0; kb < 16; ++kb)
        sf = wmma4(ldsA(xq, LDP, kb * 4, n16, kh),
                   ldsA(xk, LDP, kb * 4, n16, kh), sf);  // B = xk^T == A-read of xk
      #pragma unroll
      for (int i = 0; i < 8; ++i) {
        int m = i + 8 * kh;
        s.M[m * LDM + n16] = (m >= n16) ? (-et[m * LDE + n16] * (sf[i] + 1.0f)) : 0.f;
      }
    }
    __syncthreads();        // B2

    // ---- LN-bwd pass 1: stats (fused, redundant per thread) + gxh + partials ----
    const int rr = tid & 15, ch = tid >> 4;
    float mu, rs;
    {
      float sv = s.RS[rr] + s.RS[16 + rr] + s.RS[32 + rr] + s.RS[48 + rr];
      float qv = s.RQ[rr] + s.RQ[16 + rr] + s.RQ[32 + rr] + s.RQ[48 + rr];
      mu = sv * (1.0f / 64.0f);
      float var = qv * (1.0f / 64.0f) - mu * mu;
      rs = rsqrtf(var + 1e-6f);
      float p1 = 0.f, p2 = 0.f;
      #pragma unroll
      for (int j = 0; j < 8; ++j) {
        int cc = ch * 8 + j;
        float xh = (s.Z[rr * LDP + cc] - mu) * rs;
        float y  = s.gam[cc] * xh + s.bet[cc];
        float gx = (y - (xv[rr * LDP + cc] - xk[rr * LDP + cc])) * s.gam[cc];
        p1 += gx; p2 += gx * xh;
        s.G[rr * LDP + cc] = gx;
      }
      s.P1[rr * 8 + ch] = p1;
      s.P2[rr * 8 + ch] = p2;
      if (tid < 16) s.negEta[tid] = -et[15 * LDE + tid];
    }
    __syncthreads();        // B3

    // ---- LN-bwd pass 2: g = (D*gxh - s1 - xh*s2) / (D*std)  (s1/s2 fused) ----
    {
      float a1 = 0.f, a2 = 0.f;
      #pragma unroll
      for (int j = 0; j < 8; ++j) { a1 += s.P1[rr * 8 + j]; a2 += s.P2[rr * 8 + j]; }
      #pragma unroll
      for (int j = 0; j < 8; ++j) {
        int cc = ch * 8 + j;
        float xh = (s.Z[rr * LDP + cc] - mu) * rs;
        float gx = s.G[rr * LDP + cc];
        s.G[rr * LDP + cc] = (64.f * gx - a1 - xh * a2) * (rs * (1.0f / 64.0f));
      }
    }
    __syncthreads();        // B4

    // ---- Z1_bar = xq@W1 + b1 + M@g ----
    {
      float bb = s.b1[cb + n16];
      #pragma unroll
      for (int i = 0; i < 8; ++i) c[i] = bb;
      #pragma unroll
      for (int kb = 0; kb < 16; ++kb)
        c = wmma4(ldsA(xq, LDP, kb * 4, n16, kh),
                  ldsB(s.W1, LDW, kb * 4, cb, n16, kh), c);
      #pragma unroll
      for (int kb = 0; kb < 4; ++kb)
        c = wmma4(ldsA(s.M, LDM, kb * 4, n16, kh),
                  ldsB(s.G, LDP, kb * 4, cb, n16, kh), c);
      #pragma unroll
      for (int i = 0; i < 8; ++i) {
        int m = i + 8 * kh;
        float sv = redrow16(c[i]);
        float qv = redrow16(c[i] * c[i]);
        if (n16 == 0) { s.RS[wv * 16 + m] = sv; s.RQ[wv * 16 + m] = qv; }
      }
    }
    __syncthreads();        // B5
    if (tid < 16) {
      float sv = s.RS[tid] + s.RS[16 + tid] + s.RS[32 + tid] + s.RS[48 + tid];
      float qv = s.RQ[tid] + s.RQ[16 + tid] + s.RQ[32 + tid] + s.RQ[48 + tid];
      float m2  = sv * (1.0f / 64.0f);
      float var = qv * (1.0f / 64.0f) - m2 * m2;
      s.mu[tid]   = m2;
      s.rstd[tid] = rsqrtf(var + 1e-6f);
    }
    __syncthreads();        // B6

    // ---- out = xq + ln_fwd(Z1_bar), straight from fragment ----
    {
      float* orow = outB + (size_t)(t * 16) * 512;
      #pragma unroll
      for (int i = 0; i < 8; ++i) {
        int m  = i + 8 * kh;
        int cc = cb + n16;
        float xh = (c[i] - s.mu[m]) * s.rstd[m];
        orow[(size_t)m * 512 + cc] = xq[m * LDP + cc] + s.gam[cc] * xh + s.bet[cc];
      }
    }
    // ---- state update: W1 += (-eta15 ∘ xk)^T @ g ; b1 -= sum_k eta15[k]*g[k,:] ----
    // (B5/B6 guarantee all W1 reads of this step completed)
    #pragma unroll
    for (int rt = 0; rt < 4; ++rt) {
      int rb = rt * 16;
      v8f cw;
      #pragma unroll
      for (int i = 0; i < 8; ++i)
        cw[i] = s.W1[(rb + i + 8 * kh) * LDW + cb + n16];
      #pragma unroll
      for (int kb = 0; kb < 4; ++kb) {
        int k0 = kb * 4, k1 = k0 + 2 * kh;
        v2f a;   // A[d][k] = -eta15[k] * xk[k][d], lane row d = rb + n16
        a.x = s.negEta[k1]     * xk[k1 * LDP       + rb + n16];
        a.y = s.negEta[k1 + 1] * xk[(k1 + 1) * LDP + rb + n16];
        cw = wmma4(a, ldsB(s.G, LDP, k0, cb, n16, kh), cw);
      }
      #pragma unroll
      for (int i = 0; i < 8; ++i)
        s.W1[(rb + i + 8 * kh) * LDW + cb + n16] = cw[i];
    }
    if (tid < 64) {
      float acc = 0.f;
      #pragma unroll
      for (int k = 0; k < 16; ++k)
        acc += et[15 * LDE + k] * s.G[k * LDP + tid];
      s.b1[tid] -= acc;
    }
    // no end barrier: next step's B1 is the rendezvous; double-buffering
    // guarantees no WAR on the tile buffers across the boundary.
  }
}

extern "C" void kernel_launch(void* const* d_in, const int* in_sizes, int n_in,
                              void* d_out, int out_size, void* d_ws, size_t ws_size,
                              hipStream_t stream) {
  const float* XQ  = (const float*)d_in[0];
  const float* XK  = (const float*)d_in[1];
  const float* XV  = (const float*)d_in[2];
  const float* eta = (const float*)d_in[3];
  const float* W1  = (const float*)d_in[4];
  const float* b1  = (const float*)d_in[5];
  const float* lnw = (const float*)d_in[6];
  const float* lnb = (const float*)d_in[7];
  float* out = (float*)d_out;
  // one workgroup per (b,h) chain: 64 WGs x 4 waves (wave32)
  ttt_scan_kernel<<<64, 128, 0, stream>>>(XQ, XK, XV, eta, W1, b1, lnw, lnb, out);
}